// RandomCutoff_65962107732534
// MI455X (gfx1250) — compile-verified
//
#include <hip/hip_runtime.h>
#include <stdint.h>

// ---------------------------------------------------------------------------
// SpecAugment-style random masking, MI455X (gfx1250).
// x: (C=16, T=25600, F=256) fp32.  out = where(mask, log(eps), x).
// Pure streaming op: 0.42 GB in + 0.42 GB out -> ~36us floor @ 23.3 TB/s.
// One wave32 per 256-float row: (c,t,block) + rect descriptor are wave-
// uniform (readfirstlane -> SMEM s_load_b128 + scalar branch); data path is
// two NT-hinted b128 loads/stores per lane. Threefry precompute -> 53KB ws.
// ---------------------------------------------------------------------------

typedef __attribute__((ext_vector_type(4))) float v4f;
typedef __attribute__((ext_vector_type(4))) int   v4i;

#define C_TOT   16
#define C_MASK  13          // C - 3 channels carry masks
#define T_DIM   25600
#define F_DIM   256
#define TSTEP   100
#define NB      (T_DIM / TSTEP)   // 256 time blocks
#define TMAX    35u
#define FMAX    30u
#define NENT    (C_MASK * NB)     // 3328 descriptors
#define HALFN   (NENT / 2)        // 1664 (threefry counter split)
#define NROWS   (C_TOT * T_DIM)   // 409,600 rows == waves
#define WPB     8                 // waves per 256-thread block
#define LOG_EPS_F (-36.04365338911715f)

// Hot kernel FIRST in the file so the disasm snippet shows its codegen.
__global__ void __launch_bounds__(256)
sa_apply_kernel(const float* __restrict__ x, float* __restrict__ out,
                const v4i* __restrict__ params) {
    int lane = threadIdx.x & 31;
    int row  = __builtin_amdgcn_readfirstlane(
                   blockIdx.x * WPB + (threadIdx.x >> 5));   // wave-uniform

    // lane covers float4 indices {lane, lane+32} of this row (64 float4/row)
    int g0 = row * 64 + lane;
    int g1 = g0 + 32;

    const v4f* __restrict__ x4 = (const v4f*)x;
    v4f v0 = __builtin_nontemporal_load(x4 + g0);
    v4f v1 = __builtin_nontemporal_load(x4 + g1);

    int c = row / T_DIM;                      // scalar
    if (c < C_MASK) {
        int t     = row - c * T_DIM;
        int blk   = t / TSTEP;                // < NB always (25600/100)
        int local = t - blk * TSTEP;
        int pidx  = __builtin_amdgcn_readfirstlane(c * NB + blk);
        v4i p = params[pidx];                 // uniform addr -> s_load_b128
        if (local >= p.x && local < p.y) {    // scalar branch: row masked?
            int fa = lane << 2;               // f base for v0: 0..127
            int fb = fa + 128;                // f base for v1: 128..255
            if (fa + 0 >= p.z && fa + 0 < p.w) v0.x = LOG_EPS_F;
            if (fa + 1 >= p.z && fa + 1 < p.w) v0.y = LOG_EPS_F;
            if (fa + 2 >= p.z && fa + 2 < p.w) v0.z = LOG_EPS_F;
            if (fa + 3 >= p.z && fa + 3 < p.w) v0.w = LOG_EPS_F;
            if (fb + 0 >= p.z && fb + 0 < p.w) v1.x = LOG_EPS_F;
            if (fb + 1 >= p.z && fb + 1 < p.w) v1.y = LOG_EPS_F;
            if (fb + 2 >= p.z && fb + 2 < p.w) v1.z = LOG_EPS_F;
            if (fb + 3 >= p.z && fb + 3 < p.w) v1.w = LOG_EPS_F;
        }
    }

    v4f* __restrict__ o4 = (v4f*)out;
    __builtin_nontemporal_store(v0, o4 + g0);
    __builtin_nontemporal_store(v1, o4 + g1);
}

// ------------------------- PRNG precompute --------------------------------

struct U2 { uint32_t x, y; };

__device__ __forceinline__ uint32_t rotl32(uint32_t v, int s) {
    return (v << s) | (v >> (32 - s));
}

// JAX threefry2x32 block cipher (5 groups of 4 rounds).
__device__ __forceinline__ U2 threefry2x32(uint32_t k0, uint32_t k1,
                                           uint32_t x0, uint32_t x1) {
    uint32_t ks2 = k0 ^ k1 ^ 0x1BD11BDAu;
    x0 += k0; x1 += k1;
#define TF_R(r) { x0 += x1; x1 = rotl32(x1, r); x1 ^= x0; }
    TF_R(13) TF_R(15) TF_R(26) TF_R(6)   x0 += k1;  x1 += ks2 + 1u;
    TF_R(17) TF_R(29) TF_R(16) TF_R(24)  x0 += ks2; x1 += k0  + 2u;
    TF_R(13) TF_R(15) TF_R(26) TF_R(6)   x0 += k0;  x1 += k1  + 3u;
    TF_R(17) TF_R(29) TF_R(16) TF_R(24)  x0 += k1;  x1 += ks2 + 4u;
    TF_R(13) TF_R(15) TF_R(26) TF_R(6)   x0 += ks2; x1 += k0  + 5u;
#undef TF_R
    U2 r; r.x = x0; r.y = x1; return r;
}

// JAX uniform [0,1): bitcast((bits>>9)|0x3f800000) - 1
__device__ __forceinline__ float bits_to_unit(uint32_t b) {
    uint32_t u = (b >> 9) | 0x3f800000u;
    return __uint_as_float(u) - 1.0f;
}

// One thread per (channel, time-block) descriptor: 3328 total.
// Writes v4i {tstart, tend, fstart, fend}; coin-flip failure => empty rects.
__global__ void sa_params_kernel(v4i* __restrict__ params) {
    int e = blockIdx.x * blockDim.x + threadIdx.x;
    if (e >= NENT) return;

    // jax.random.key(42) -> data = [0, 42]
    const uint32_t B0 = 0u, B1 = 42u;

    // split(key, 2): blocks (0,2),(1,3); kcoin=(s0.x,s1.x), kmask=(s0.y,s1.y)
    U2 s0 = threefry2x32(B0, B1, 0u, 2u);
    U2 s1 = threefry2x32(B0, B1, 1u, 3u);
    uint32_t kc0 = s0.x, kc1 = s1.x;
    uint32_t km0 = s0.y, km1 = s1.y;

    // split(kmask, 4): blocks (i, i+4); flat words -> k1..k4
    U2 m0 = threefry2x32(km0, km1, 0u, 4u);
    U2 m1 = threefry2x32(km0, km1, 1u, 5u);
    U2 m2 = threefry2x32(km0, km1, 2u, 6u);
    U2 m3 = threefry2x32(km0, km1, 3u, 7u);
    uint32_t k1a = m0.x, k1b = m1.x;   // tlen key
    uint32_t k2a = m2.x, k2b = m3.x;   // tstart key
    uint32_t k3a = m0.y, k3b = m1.y;   // flen key
    uint32_t k4a = m2.y, k4b = m3.y;   // fstart key

    // word e of a 3328-word draw: block i<1664 -> ctr (i, i+1664), .x else .y
    bool lo = (e < HALFN);
    uint32_t c0 = lo ? (uint32_t)e : (uint32_t)(e - HALFN);
    uint32_t c1 = lo ? (uint32_t)(e + HALFN) : (uint32_t)e;
    U2 w1 = threefry2x32(k1a, k1b, c0, c1);
    U2 w2 = threefry2x32(k2a, k2b, c0, c1);
    U2 w3 = threefry2x32(k3a, k3b, c0, c1);
    U2 w4 = threefry2x32(k4a, k4b, c0, c1);
    uint32_t b1 = lo ? w1.x : w1.y;
    uint32_t b2 = lo ? w2.x : w2.y;
    uint32_t b3 = lo ? w3.x : w3.y;
    uint32_t b4 = lo ? w4.x : w4.y;

    int tl = (int)(b1 % TMAX);
    int ts = (int)floorf(bits_to_unit(b2) * (float)(TSTEP - tl));
    int fl = (int)(b3 % FMAX);
    int fs = (int)floorf(bits_to_unit(b4) * (float)(F_DIM - fl));

    // coin: uniform(kcoin, ()) <= 0.5 (odd-size draw pads ctr to (0,0))
    U2 cw = threefry2x32(kc0, kc1, 0u, 0u);
    bool apply = bits_to_unit(cw.x) <= 0.5f;

    v4i p;
    p.x = ts;
    p.y = apply ? (ts + tl) : ts;   // empty interval when coin says no-mask
    p.z = fs;
    p.w = apply ? (fs + fl) : fs;
    params[e] = p;
}

extern "C" void kernel_launch(void* const* d_in, const int* in_sizes, int n_in,
                              void* d_out, int out_size, void* d_ws, size_t ws_size,
                              hipStream_t stream) {
    const float* x   = (const float*)d_in[0];
    float*       out = (float*)d_out;
    v4i*         par = (v4i*)d_ws;        // 3328 * 16 B = 53 KB of scratch

    (void)in_sizes; (void)n_in; (void)out_size; (void)ws_size;

    sa_params_kernel<<<(NENT + 255) / 256, 256, 0, stream>>>(par);

    // one wave per row: 409,600 waves / 8 waves-per-block = 51,200 blocks
    sa_apply_kernel<<<NROWS / WPB, 256, 0, stream>>>(x, out, par);
}